// TWIRLSUnfoldingAndAttention_6399501271288
// MI455X (gfx1250) — compile-verified
//
#include <hip/hip_runtime.h>

typedef __attribute__((ext_vector_type(2))) float v2f;
typedef __attribute__((ext_vector_type(8))) float v8f;

#define FD 64          // feature dim
#define TB 256         // block size (8 waves of 32)

// ---------------- simple elementwise kernels ----------------

__global__ void fill_ones_kernel(float* __restrict__ w, int E) {
    int i = blockIdx.x * blockDim.x + threadIdx.x;
    if (i < E) w[i] = 1.0f;
}

__global__ void deg_accum_kernel(const float* __restrict__ w, const int* __restrict__ dst,
                                 float* __restrict__ deg, int E) {
    int i = blockIdx.x * blockDim.x + threadIdx.x;
    if (i < E) unsafeAtomicAdd(&deg[dst[i]], w[i]);
}

// coef[e] = w[e] * deg[src[e]]^-0.5   (recomputed only when w/deg change)
__global__ void coef_kernel(const float* __restrict__ w, const float* __restrict__ deg,
                            const int* __restrict__ src, float* __restrict__ coef, int E) {
    int i = blockIdx.x * blockDim.x + threadIdx.x;
    if (i < E) coef[i] = w[i] * __frsqrt_rn(deg[src[i]]);
}

// ---------------- propagation: gather + scatter-add ----------------
// 16 threads per edge, float4 each: acc[dst] += coef[e] * Y[src]
__global__ void scatter_kernel(const float* __restrict__ Y, const float* __restrict__ coef,
                               const int* __restrict__ src, const int* __restrict__ dst,
                               float* __restrict__ acc, int E) {
    int tid = blockIdx.x * blockDim.x + threadIdx.x;
    if (tid >= E * 16) return;
    int e = tid >> 4;
    int f = (tid & 15) << 2;
    int s = src[e], d = dst[e];
    float t = coef[e];
    float4 y = *(const float4*)(Y + (size_t)s * FD + f);
    float* pa = acc + (size_t)d * FD + f;
    unsafeAtomicAdd(pa + 0, t * y.x);
    unsafeAtomicAdd(pa + 1, t * y.y);
    unsafeAtomicAdd(pa + 2, t * y.z);
    unsafeAtomicAdd(pa + 3, t * y.w);
}

// Y = 0.5*Y + 0.5*rsqrt(deg)*acc + 0.5*X/deg   (alp=0.5, lam=1)
__global__ void update_kernel(float* __restrict__ Y, const float* __restrict__ acc,
                              const float* __restrict__ X, const float* __restrict__ deg, int N) {
    int tid = blockIdx.x * blockDim.x + threadIdx.x;
    if (tid >= N * 16) return;
    int v = tid >> 4;
    int f = (tid & 15) << 2;
    float dg  = deg[v];
    float rs  = __frsqrt_rn(dg);
    float inv = 1.0f / dg;
    size_t off = (size_t)v * FD + f;
    float4 y = *(const float4*)(Y + off);
    float4 a = *(const float4*)(acc + off);
    float4 x = *(const float4*)(X + off);
    y.x = 0.5f * y.x + 0.5f * rs * a.x + 0.5f * inv * x.x;
    y.y = 0.5f * y.y + 0.5f * rs * a.y + 0.5f * inv * x.y;
    y.z = 0.5f * y.z + 0.5f * rs * a.z + 0.5f * inv * x.z;
    y.w = 0.5f * y.w + 0.5f * rs * a.w + 0.5f * inv * x.w;
    *(float4*)(Y + off) = y;
}

// ---------------- attention ----------------
// h_norm[v] = sum(Y[v]^2); 16 lanes per node, float4 each, shfl_xor reduce
__global__ void hnorm_kernel(const float* __restrict__ Y, float* __restrict__ hn, int N) {
    int tid = blockIdx.x * blockDim.x + threadIdx.x;
    int node = tid >> 4;
    float s = 0.0f;
    if (node < N) {
        int f = (tid & 15) << 2;
        float4 y = *(const float4*)(Y + (size_t)node * FD + f);
        s = y.x * y.x + y.y * y.y + y.z * y.z + y.w * y.w;
    }
    s += __shfl_xor(s, 1, 32);
    s += __shfl_xor(s, 2, 32);
    s += __shfl_xor(s, 4, 32);
    s += __shfl_xor(s, 8, 32);
    if (node < N && (tid & 15) == 0) hn[node] = s;
}

// One wave handles 16 edges. Exact-f32 WMMA (16x16x4) computes the 16x16 cross
// Gram matrix of Y[src] rows vs Y[dst] rows over D=64 (16 chained K-chunks);
// diagonal = per-edge dot products. Then w[e] = 1/max(sqrt(relu(d2)+1e-7),tau)+1e-9
// and weighted degree is accumulated.
__global__ void attn_kernel(const float* __restrict__ Y, const float* __restrict__ hn,
                            const int* __restrict__ src, const int* __restrict__ dst,
                            float* __restrict__ w, float* __restrict__ deg, int E) {
    int wid  = (blockIdx.x * blockDim.x + threadIdx.x) >> 5;   // global wave id
    int lane = threadIdx.x & 31;
    int e0 = wid * 16;
    if (e0 >= E) return;                 // wave-uniform: EXEC stays all-ones for WMMA

    // Per ISA 16x4 f32 layout: lane g*16+i holds row i, cols {k0+2g, k0+2g+1}
    int i = lane & 15;
    int g = lane >> 4;
    int el = e0 + i; el = (el < E) ? el : (E - 1);   // clamp tail (select, no branch)
    int s = src[el], d = dst[el];
    const float* ps = Y + (size_t)s * FD + 2 * g;    // A: src rows
    const float* pd = Y + (size_t)d * FD + 2 * g;    // B: dst rows (columns of B)

    v8f c = {};
#pragma unroll
    for (int kc = 0; kc < 16; ++kc) {                // K = 64 = 16 chunks of 4
        v2f a, b;
        a.x = ps[4 * kc];  a.y = ps[4 * kc + 1];
        b.x = pd[4 * kc];  b.y = pd[4 * kc + 1];
        c = __builtin_amdgcn_wmma_f32_16x16x4_f32(false, a, false, b,
                                                  (short)0, c, false, false);
    }

    // Diagonal (i,i): i<8 -> lane i, VGPR i; i>=8 -> lane i+16, VGPR i-8
    bool own = (lane < 8) || (lane >= 24);
    int  ei  = (lane < 8) ? lane : (lane - 16);
    int  e   = e0 + ei;
    if (own && e < E) {
        int idx = lane & 7;
        float dot = c[0];
#pragma unroll
        for (int j = 1; j < 8; ++j) dot = (idx == j) ? c[j] : dot;
        int ss = src[e], dd = dst[e];
        float we = hn[ss] + hn[dd] - 2.0f * dot;     // squared distance
        we = fmaxf(we, 0.0f) + 1e-7f;
        we = sqrtf(we);                              // pow(we, 1 - 0.5*P), P=1
        we = fmaxf(we, 0.2f);                        // tau
        float wv = 1.0f / we + 1e-9f;
        w[e] = wv;
        unsafeAtomicAdd(&deg[dd], wv);
    }
}

// ---------------- driver ----------------

extern "C" void kernel_launch(void* const* d_in, const int* in_sizes, int n_in,
                              void* d_out, int out_size, void* d_ws, size_t ws_size,
                              hipStream_t stream) {
    const float* X  = (const float*)d_in[0];
    const int* src  = (const int*)d_in[1];
    const int* dst  = (const int*)d_in[2];
    int ND = in_sizes[0];
    int N  = ND / FD;
    int E  = in_sizes[1];

    float* Y    = (float*)d_out;           // Y lives in d_out
    float* acc  = (float*)d_ws;            // N*FD
    float* wgt  = acc  + (size_t)N * FD;   // E
    float* coef = wgt  + E;                // E
    float* deg  = coef + E;                // N
    float* hn   = deg  + N;                // N

    int gE  = (E + TB - 1) / TB;
    int gE16 = (int)(((long long)E * 16 + TB - 1) / TB);
    int gN16 = (N * 16 + TB - 1) / TB;
    int gAttn = (E + 127) / 128;           // 16 edges/wave, 8 waves/block

    // Y = X; w = 1; deg = in-degree; coef = w * rsqrt(deg[src])
    hipMemcpyAsync(Y, X, (size_t)ND * sizeof(float), hipMemcpyDeviceToDevice, stream);
    fill_ones_kernel<<<gE, TB, 0, stream>>>(wgt, E);
    hipMemsetAsync(deg, 0, (size_t)N * sizeof(float), stream);
    deg_accum_kernel<<<gE, TB, 0, stream>>>(wgt, dst, deg, E);
    coef_kernel<<<gE, TB, 0, stream>>>(wgt, deg, src, coef, E);

    for (int k = 0; k < 4; ++k) {
        hipMemsetAsync(acc, 0, (size_t)ND * sizeof(float), stream);
        scatter_kernel<<<gE16, TB, 0, stream>>>(Y, coef, src, dst, acc, E);
        update_kernel<<<gN16, TB, 0, stream>>>(Y, acc, X, deg, N);

        if (k == 1) {   // attention after 2nd propagation step
            hnorm_kernel<<<gN16, TB, 0, stream>>>(Y, hn, N);
            hipMemsetAsync(deg, 0, (size_t)N * sizeof(float), stream);
            attn_kernel<<<gAttn, TB, 0, stream>>>(Y, hn, src, dst, wgt, deg, E);
            coef_kernel<<<gE, TB, 0, stream>>>(wgt, deg, src, coef, E);
        }
    }
}